// EmbeddingSharded_69810398429479
// MI455X (gfx1250) — compile-verified
//
#include <hip/hip_runtime.h>

// Embedding gather: out[r, :] = W[x[r], :] + b[:]
// x: [ROWS] int token ids, W: [VOCAB, D_MODEL] f32, b: [D_MODEL] f32,
// out: [ROWS, D_MODEL] f32.  D_MODEL = 2048, ROWS = 8192.
//
// Pure bandwidth problem (~128 MiB traffic -> ~5.5 us at 23.3 TB/s).
// One block per row; b128 vector loads/stores; NT store hints for the
// streaming output; gfx1250 global_prefetch for the next gathered row.

#define D_MODEL 2048

typedef float v4f __attribute__((ext_vector_type(4)));

__global__ __launch_bounds__(256) void embed_gather_kernel(
    const int*   __restrict__ x,    // [rows] token ids
    const float* __restrict__ W,    // [VOCAB, D_MODEL]
    const float* __restrict__ bias, // [D_MODEL]
    float*       __restrict__ out,  // [rows, D_MODEL]
    int rows)
{
    const int row = blockIdx.x;
    if (row >= rows) return;

    // Token id is block-uniform; compiler can scalarize this load.
    const int tok = x[row];

    const v4f* __restrict__ src = (const v4f*)(W + (size_t)tok * D_MODEL);
    const v4f* __restrict__ bv  = (const v4f*)bias;
    v4f*       __restrict__ dst = (v4f*)(out + (size_t)row * D_MODEL);

    // Prefetch the NEXT row's embedding into cache while this row streams.
    // 8 KiB row = 64 x 128B cachelines; lanes 0..63 each touch one line.
    // (gfx1250: lowers to global_prefetch_b8 — no VGPR return, no counter.)
    if (row + 1 < rows && threadIdx.x < 64) {
        const int tok_next = x[row + 1];
        const float* pf = W + (size_t)tok_next * D_MODEL + (size_t)threadIdx.x * 32;
        __builtin_prefetch(pf, 0, 1);
    }

    // 2048 floats = 512 float4; 256 threads -> 2 fully-coalesced iterations.
    const int nvec = D_MODEL / 4; // 512
    #pragma unroll
    for (int i = 0; i < 2; ++i) {
        const int c = threadIdx.x + i * 256;
        if (c < nvec) {
            v4f w = src[c];
            v4f bb = bv[c];
            v4f r = w + bb;
            // Output is written once and never re-read by this dispatch:
            // non-temporal store keeps L2 free for gathered W rows.
            __builtin_nontemporal_store(r, &dst[c]);
        }
    }
}

extern "C" void kernel_launch(void* const* d_in, const int* in_sizes, int n_in,
                              void* d_out, int out_size, void* d_ws, size_t ws_size,
                              hipStream_t stream) {
    (void)n_in; (void)d_ws; (void)ws_size;

    const int*   x    = (const int*)d_in[0];    // [rows] token ids (int)
    const float* W    = (const float*)d_in[1];  // [VOCAB, D_MODEL] f32
    const float* bias = (const float*)d_in[2];  // [D_MODEL] f32
    float*       out  = (float*)d_out;          // [rows, D_MODEL] f32

    const int rows = in_sizes[0];               // 4 * 2048 = 8192
    (void)out_size;

    dim3 grid((unsigned)rows);
    dim3 block(256);
    embed_gather_kernel<<<grid, block, 0, stream>>>(x, W, bias, out, rows);
}